// InterAmazon_76879914598412
// MI455X (gfx1250) — compile-verified
//
#include <hip/hip_runtime.h>

// ---------------------------------------------------------------------------
// GraphSAGE layer fused kernel for MI455X (gfx1250, wave32, WMMA).
//   out[e,b] = relu( sum_k W[e,k] * combined[b,k] )
//   combined[b, 0:256]   = features[nodes[b]]
//   combined[b, 256:512] = mean_j features[neigh_idx[b,j]]
// Memory-bound (~250MB @ 23.3 TB/s); fp32 kept end-to-end via
// V_WMMA_F32_16X16X4_F32 (exact-precision matrix pipe).
// ---------------------------------------------------------------------------

typedef __attribute__((ext_vector_type(2))) float v2f;
typedef __attribute__((ext_vector_type(4))) float v4f;
typedef __attribute__((ext_vector_type(8))) float v8f;

#define FEAT       256
#define EMBED      256
#define BATCH      16384
#define NSAMP      10
#define KDIM       (2 * FEAT)        // 512
#define BT         64                // batch columns per workgroup
#define LDS_STRIDE (KDIM + 4)        // 516 floats/row: pad kills LDS bank conflicts
#define THREADS    256               // 8 wave32

__global__ __launch_bounds__(THREADS)
void sage_fused_wmma(const float* __restrict__ feat,
                     const float* __restrict__ W,
                     const int*   __restrict__ nodes,
                     const int*   __restrict__ nidx,
                     float*       __restrict__ out)
{
    extern __shared__ float comb[];          // [BT][LDS_STRIDE] fp32, 132,096 B
    const int t   = threadIdx.x;
    const int wgB = blockIdx.x * BT;         // first batch column of this WG

    // ---------------- Phase 1: gather self + neighbor-mean into LDS --------
    {
        const int bl  = t & (BT - 1);        // local batch row 0..63
        const int seg = t >> 6;              // 64-float segment 0..3
        const int bg  = wgB + bl;

        const int self = nodes[bg];
        int nb[NSAMP];
#pragma unroll
        for (int j = 0; j < NSAMP; ++j) nb[j] = nidx[bg * NSAMP + j];

        const v4f* f4       = (const v4f*)feat;      // feature row = 64 float4
        float*     selfRow  = comb + bl * LDS_STRIDE + seg * 64;
        float*     neighRow = selfRow + FEAT;
        const long fbase    = (long)seg * 16;

#pragma unroll 4
        for (int c = 0; c < 16; ++c) {
            v4f sv  = f4[(long)self * 64 + fbase + c];
            v4f acc = f4[(long)nb[0] * 64 + fbase + c];
#pragma unroll
            for (int j = 1; j < NSAMP; ++j)
                acc += f4[(long)nb[j] * 64 + fbase + c];
            *(v4f*)(selfRow  + c * 4) = sv;
            *(v4f*)(neighRow + c * 4) = acc * 0.1f;   // mean over 10 samples
        }
    }
    __syncthreads();

    // ---------------- Phase 2: WMMA GEMM + fused ReLU -----------------------
    // Wave w owns E-tiles {2w, 2w+1} x all 4 B-tiles of this workgroup.
    const int lane   = t & 31;
    const int wave   = t >> 5;
    const int half   = lane >> 4;            // 0 -> K{0,1}, 1 -> K{2,3} (ISA A/B f32 layout)
    const int lane16 = lane & 15;

#pragma unroll
    for (int eo = 0; eo < 2; ++eo) {
        const int et = wave * 2 + eo;                     // E-tile 0..15
        // A fragment source: W row (et*16 + lane16), cols k0 + 2*half + {0,1}
        const float* wrow = W + (long)(et * 16 + lane16) * KDIM + half * 2;
        // B fragment source: comb row (bt*16 + lane16), cols k0 + 2*half + {0,1}
        const float* crow = comb + lane16 * LDS_STRIDE + half * 2;

        v8f acc0 = {}, acc1 = {}, acc2 = {}, acc3 = {};

#pragma unroll 2
        for (int k0 = 0; k0 < KDIM; k0 += 4) {
            v2f a  = *(const v2f*)(wrow + k0);                       // global (L2-hot)
            v2f b0 = *(const v2f*)(crow + 0 * 16 * LDS_STRIDE + k0); // ds_load_b64
            v2f b1 = *(const v2f*)(crow + 1 * 16 * LDS_STRIDE + k0);
            v2f b2 = *(const v2f*)(crow + 2 * 16 * LDS_STRIDE + k0);
            v2f b3 = *(const v2f*)(crow + 3 * 16 * LDS_STRIDE + k0);
            acc0 = __builtin_amdgcn_wmma_f32_16x16x4_f32(false, a, false, b0, (short)0, acc0, false, false);
            acc1 = __builtin_amdgcn_wmma_f32_16x16x4_f32(false, a, false, b1, (short)0, acc1, false, false);
            acc2 = __builtin_amdgcn_wmma_f32_16x16x4_f32(false, a, false, b2, (short)0, acc2, false, false);
            acc3 = __builtin_amdgcn_wmma_f32_16x16x4_f32(false, a, false, b3, (short)0, acc3, false, false);
        }

        // D layout (32-bit C/D 16x16): vgpr r -> M=r (lanes 0-15) / M=r+8 (lanes 16-31),
        // N = lane16. ReLU fused into the stores.
        const int ebase = et * 16 + half * 8;
#pragma unroll
        for (int r = 0; r < 8; ++r) {
            const long erow = ebase + r;
            float* o = out + erow * BATCH + wgB + lane16;
            o[0 * 16] = fmaxf(acc0[r], 0.0f);
            o[1 * 16] = fmaxf(acc1[r], 0.0f);
            o[2 * 16] = fmaxf(acc2[r], 0.0f);
            o[3 * 16] = fmaxf(acc3[r], 0.0f);
        }
    }
}

extern "C" void kernel_launch(void* const* d_in, const int* in_sizes, int n_in,
                              void* d_out, int out_size, void* d_ws, size_t ws_size,
                              hipStream_t stream) {
    (void)in_sizes; (void)n_in; (void)d_ws; (void)ws_size; (void)out_size;
    const float* feat  = (const float*)d_in[0];   // [1e6, 256] fp32
    const float* W     = (const float*)d_in[1];   // [256, 512] fp32
    const int*   nodes = (const int*)d_in[2];     // [16384] int32
    const int*   nidx  = (const int*)d_in[3];     // [16384, 10] int32
    float*       out   = (float*)d_out;           // [256, 16384] fp32

    const int    grid     = BATCH / BT;                       // 256 workgroups
    const size_t ldsBytes = (size_t)BT * LDS_STRIDE * sizeof(float); // 132,096 B (<320KB/WGP)

    sage_fused_wmma<<<grid, THREADS, ldsBytes, stream>>>(feat, W, nodes, nidx, out);
}